// Conv2D_20744692039894
// MI455X (gfx1250) — compile-verified
//
#include <hip/hip_runtime.h>

typedef __attribute__((ext_vector_type(2))) float v2f;
typedef __attribute__((ext_vector_type(8))) float v8f;

#define CIN   8
#define COUT  16
#define HDIM  56
#define WDIM  56
#define SDIM  28
#define NDIM  (SDIM * SDIM)       // 784 symbolic coeffs per pixel
#define KDIM  (CIN * 9)           // 72 reduction length
#define NTILE (NDIM / 16)         // 49 N-tiles of 16
#define NPAIR 25                  // ceil(49/2); pair 24 duplicates tile 48
#define PIX   (HDIM * WDIM)       // 3136
#define CSTR  (HDIM * WDIM * NDIM)            // 2,458,624 : ci stride in floats
#define SYM_ELEMS ((long)COUT * PIX * NDIM)   // 39,337,984 per sym output

// One wave per (pixel, n-tile-pair). K is reordered as k = tap*8 + ci so each
// WMMA k-step of 4 stays inside one (kh,kw) tap: tap geometry and the halo
// validity check are hoisted to compile-time-unrolled tap level, and the
// validity branch is made wave-uniform (scalar) via readfirstlane.
__global__ __launch_bounds__(256) void ibp_conv_sym_kernel(
    const float* __restrict__ lx, const float* __restrict__ ux,
    const float* __restrict__ wt,
    float* __restrict__ lxo, float* __restrict__ uxo)
{
    const int lane = threadIdx.x & 31;
    const int wave = blockIdx.x * (blockDim.x >> 5) + (threadIdx.x >> 5);

    const int pr  = wave % NPAIR;           // n-tile pair index
    const int pix = wave / NPAIR;           // < 3136 (exact grid, no guard)
    const int wq  = pix % WDIM;
    const int hq  = pix / WDIM;

    const int hi  = lane >> 4;              // selects K sub-pair {0,1} vs {2,3}
    const int nl  = lane & 15;
    const int ci0 = 2 * hi;                 // base ci within a k-step
    // lane's n positions for the two tiles (pair 24: both = tile 48)
    const unsigned n0 = (unsigned)(pr * 32) + nl;
    const unsigned n1 = (unsigned)min(pr * 32 + 16, 768) + nl;

    // Preload this lane's A-matrix entries, reordered k = tap*8 + ci.
    // wr[tap*2 + s][j] = W[co = nl][ci = s*4 + ci0 + j][tap], stride 9 in k.
    v2f wr[18];
    const float* wbase = wt + nl * KDIM;
    #pragma unroll
    for (int tap = 0; tap < 9; ++tap) {
        #pragma unroll
        for (int s = 0; s < 2; ++s) {
            wr[tap * 2 + s][0] = wbase[(s * 4 + ci0 + 0) * 9 + tap];
            wr[tap * 2 + s][1] = wbase[(s * 4 + ci0 + 1) * 9 + tap];
        }
    }

    v8f accl0 = {}, accl1 = {};   // lower bound, tiles 0/1
    v8f accu0 = {}, accu1 = {};   // upper bound, tiles 0/1

    #pragma unroll
    for (int tap = 0; tap < 9; ++tap) {
        const int kh = tap / 3;             // compile-time constants
        const int kw = tap % 3;
        const int y  = hq + kh - 1;         // pad = 1
        const int x  = wq + kw - 1;
        const int valid = (int)(((unsigned)y < HDIM) & ((unsigned)x < WDIM));

        v2f bl[2][2] = {}, bu[2][2] = {};   // [kstep s][tile]
        if (__builtin_amdgcn_readfirstlane(valid)) {   // wave-uniform branch
            const unsigned row = (unsigned)(y * WDIM + x) * NDIM;
            #pragma unroll
            for (int s = 0; s < 2; ++s) {
                const unsigned o = row + (unsigned)(s * 4 + ci0) * CSTR;
                bl[s][0][0] = lx[o + n0];        bl[s][0][1] = lx[o + CSTR + n0];
                bl[s][1][0] = lx[o + n1];        bl[s][1][1] = lx[o + CSTR + n1];
                bu[s][0][0] = ux[o + n0];        bu[s][0][1] = ux[o + CSTR + n0];
                bu[s][1][0] = ux[o + n1];        bu[s][1][1] = ux[o + CSTR + n1];
            }
        }

        #pragma unroll
        for (int s = 0; s < 2; ++s) {
            const v2f w = wr[tap * 2 + s];
            v2f wp, wn;                      // sign-split in registers
            wp[0] = fmaxf(w[0], 0.0f); wp[1] = fmaxf(w[1], 0.0f);
            wn[0] = fminf(w[0], 0.0f); wn[1] = fminf(w[1], 0.0f);

            accl0 = __builtin_amdgcn_wmma_f32_16x16x4_f32(false, wp, false, bl[s][0], (short)0, accl0, false, false);
            accl0 = __builtin_amdgcn_wmma_f32_16x16x4_f32(false, wn, false, bu[s][0], (short)0, accl0, false, false);
            accu0 = __builtin_amdgcn_wmma_f32_16x16x4_f32(false, wp, false, bu[s][0], (short)0, accu0, false, false);
            accu0 = __builtin_amdgcn_wmma_f32_16x16x4_f32(false, wn, false, bl[s][0], (short)0, accu0, false, false);

            accl1 = __builtin_amdgcn_wmma_f32_16x16x4_f32(false, wp, false, bl[s][1], (short)0, accl1, false, false);
            accl1 = __builtin_amdgcn_wmma_f32_16x16x4_f32(false, wn, false, bu[s][1], (short)0, accl1, false, false);
            accu1 = __builtin_amdgcn_wmma_f32_16x16x4_f32(false, wp, false, bu[s][1], (short)0, accu1, false, false);
            accu1 = __builtin_amdgcn_wmma_f32_16x16x4_f32(false, wn, false, bl[s][1], (short)0, accu1, false, false);
        }
    }

    // C/D layout: acc[r] = D[M = r + 8*hi][N = lane&15]; 16-float contiguous
    // stores per half-wave. Streamed write-once -> non-temporal (keep L2 for
    // the 157MB input working set that is re-read across the 9 taps).
    const unsigned pixoff = (unsigned)(hq * WDIM + wq) * NDIM;
    #pragma unroll
    for (int r = 0; r < 8; ++r) {
        const unsigned m    = (unsigned)(r + 8 * hi);         // output channel
        const unsigned base = m * (unsigned)(PIX) * NDIM + pixoff;
        __builtin_nontemporal_store(accl0[r], lxo + base + n0);
        __builtin_nontemporal_store(accl1[r], lxo + base + n1);
        __builtin_nontemporal_store(accu0[r], uxo + base + n0);
        __builtin_nontemporal_store(accu1[r], uxo + base + n1);
    }
}

// Tiny constant-offset conv (+bias): 16*56*56 outputs, scalar is fine.
__global__ __launch_bounds__(256) void ibp_conv_const_kernel(
    const float* __restrict__ lc, const float* __restrict__ uc,
    const float* __restrict__ wt, const float* __restrict__ bias,
    float* __restrict__ lco, float* __restrict__ uco)
{
    const int idx = blockIdx.x * blockDim.x + threadIdx.x;   // co*PIX + pix, exact
    const int co  = idx / PIX;
    const int pix = idx - co * PIX;
    const int wq  = pix % WDIM;
    const int hq  = pix / WDIM;

    float al = bias[co];
    float au = bias[co];
    for (int ci = 0; ci < CIN; ++ci) {
        for (int kh = 0; kh < 3; ++kh) {
            const int y = hq + kh - 1;
            if ((unsigned)y >= HDIM) continue;
            for (int kw = 0; kw < 3; ++kw) {
                const int x = wq + kw - 1;
                if ((unsigned)x >= WDIM) continue;
                const float wv = wt[((co * CIN + ci) * 3 + kh) * 3 + kw];
                const float wp = fmaxf(wv, 0.0f);
                const float wn = fminf(wv, 0.0f);
                const float l  = lc[(ci * HDIM + y) * WDIM + x];
                const float u  = uc[(ci * HDIM + y) * WDIM + x];
                al += wp * l + wn * u;
                au += wp * u + wn * l;
            }
        }
    }
    lco[idx] = al;
    uco[idx] = au;
}

extern "C" void kernel_launch(void* const* d_in, const int* in_sizes, int n_in,
                              void* d_out, int out_size, void* d_ws, size_t ws_size,
                              hipStream_t stream) {
    (void)in_sizes; (void)n_in; (void)out_size; (void)d_ws; (void)ws_size;

    const float* lx = (const float*)d_in[0];   // (8,56,56,28,28)
    const float* ux = (const float*)d_in[1];   // (8,56,56,28,28)
    const float* lc = (const float*)d_in[2];   // (8,56,56)
    const float* uc = (const float*)d_in[3];   // (8,56,56)
    const float* wt = (const float*)d_in[4];   // (16,8,3,3)
    const float* bs = (const float*)d_in[5];   // (16,)

    float* out = (float*)d_out;
    float* lxo = out;                                   // (16,56,56,28,28)
    float* uxo = out + SYM_ELEMS;                       // (16,56,56,28,28)
    float* lco = out + 2 * SYM_ELEMS;                   // (16,56,56)
    float* uco = lco + (long)COUT * PIX;                // (16,56,56)

    // 3136 pixels * 25 tile-pairs = 78,400 waves; 8 waves/block -> 9,800 blocks.
    const int total_waves = PIX * NPAIR;
    hipLaunchKernelGGL(ibp_conv_sym_kernel,
                       dim3(total_waves / 8), dim3(256), 0, stream,
                       lx, ux, wt, lxo, uxo);

    // 16*3136 = 50,176 outputs = 196 blocks * 256 threads, exact.
    hipLaunchKernelGGL(ibp_conv_const_kernel,
                       dim3((COUT * PIX) / 256), dim3(256), 0, stream,
                       lc, uc, wt, bs, lco, uco);
}